// Matcher_16406775071019
// MI455X (gfx1250) — compile-verified
//
#include <hip/hip_runtime.h>
#include <hip/hip_bf16.h>
#include <math.h>

// Problem constants (match reference)
#define B_ 8
#define Q_ 2048
#define G_ 128
#define C_ 512

typedef __attribute__((ext_vector_type(2))) float v2f;
typedef __attribute__((ext_vector_type(8))) float v8f;

// 512 + 4 floats of pad: row stride = 516 -> rows land 4 banks apart,
// so the 16 same-column reads in the A-fragment load hit 16 distinct banks.
#define LDS_STRIDE 516

// ---------------------------------------------------------------------------
// Kernel 1: fused focal-diff + one-hot GEMM (WMMA f32 16x16x4) cost builder.
// One block = one (batch, 16-row q-tile). 8 waves, one 16-col g-tile each.
//   D = diff(16x512) x (W_CLASS * onehot(512x16)) + [5*center + size - 2*gious]
// Exact in f32: each dot product has exactly one nonzero term.
// Writes final_cost [B,Q,G] to output and a transposed copy [B,G,Q] to ws.
// ---------------------------------------------------------------------------
__global__ __launch_bounds__(256)
void cost_wmma_kernel(const float* __restrict__ sem,
                      const float* __restrict__ center,
                      const float* __restrict__ size_,
                      const float* __restrict__ gious,
                      const int*   __restrict__ labels,
                      float* __restrict__ fc_out,   // [B,Q,G]
                      float* __restrict__ fcT)      // [B,G,Q]
{
    __shared__ float diffs[16 * LDS_STRIDE];

    const int qt  = blockIdx.x;          // 0..Q/16-1
    const int b   = blockIdx.y;          // 0..B-1
    const int qbase = qt * 16;
    const int tid = threadIdx.x;

    // Phase 1: focal diff for 16 q-rows x 512 classes -> LDS
    for (int e = tid; e < 16 * C_; e += 256) {
        int row = e >> 9;                // /512
        int c   = e & (C_ - 1);
        float x = sem[((size_t)b * Q_ + qbase + row) * C_ + c];
        float p = 1.0f / (1.0f + __expf(-x));
        // neg = (1-ALPHA) * p^2 * -log1p(-(p-1e-8)) = 0.75*p^2*-log(1-p+1e-8)
        float neg = 0.75f * p * p * (-__logf(1.0f - p + 1e-8f));
        float pos = 0.25f * (1.0f - p) * (1.0f - p) * (-__logf(p + 1e-8f));
        diffs[row * LDS_STRIDE + c] = pos - neg;
    }
    __syncthreads();

    // Phase 2: one wave per g-tile
    const int wave  = tid >> 5;
    const int lane  = tid & 31;
    const int gbase = wave * 16;
    const int half  = lane >> 4;         // 0: lanes 0-15, 1: lanes 16-31
    const int l16   = lane & 15;

    const int lab = labels[b * G_ + gbase + l16];  // N-column owned by lane

    // Accumulator C preloaded with the non-class weighted terms.
    // C/D layout: VGPR r -> M = r + 8*half, N = l16.
    v8f acc;
    #pragma unroll
    for (int r = 0; r < 8; ++r) {
        int row = r + 8 * half;
        size_t idx = ((size_t)b * Q_ + qbase + row) * G_ + gbase + l16;
        acc[r] = 5.0f * center[idx] + size_[idx] - 2.0f * gious[idx];
    }

    // K loop: A 16x4 f32 fragment (M=l16, K = k+2*half, k+2*half+1),
    //         B 4x16 one-hot fragment scaled by W_CLASS=2.0, same K mapping.
    for (int k = 0; k < C_; k += 4) {
        int kk = k + 2 * half;
        v2f a, bm;
        a[0] = diffs[l16 * LDS_STRIDE + kk];
        a[1] = diffs[l16 * LDS_STRIDE + kk + 1];
        bm[0] = (lab == kk)     ? 2.0f : 0.0f;
        bm[1] = (lab == kk + 1) ? 2.0f : 0.0f;
        acc = __builtin_amdgcn_wmma_f32_16x16x4_f32(
            /*neg_a=*/false, a, /*neg_b=*/false, bm,
            /*c_mod=*/(short)0, acc, /*reuse_a=*/false, /*reuse_b=*/false);
    }

    // Epilogue: D is final_cost tile; write row-major + transposed copy.
    #pragma unroll
    for (int r = 0; r < 8; ++r) {
        int row = r + 8 * half;
        int q = qbase + row, g = gbase + l16;
        float val = acc[r];
        fc_out[((size_t)b * Q_ + q) * G_ + g] = val;
        fcT[((size_t)b * G_ + g) * Q_ + q]    = val;
    }
}

// ---------------------------------------------------------------------------
// Kernel 2: reference assignment solve, one block per batch.
// Mirrors the reference exactly: minv/way copies are never written back, so
// the running-min state stays infinite, way stays 0, and augmentation
// degenerates to p[j_final] = i. Dual updates happen on EVERY inner
// iteration (including the last, before the break), argmin ties pick the
// smallest column index (numpy first-occurrence semantics).
// ---------------------------------------------------------------------------
__global__ __launch_bounds__(1024)
void hungarian_kernel(const float* __restrict__ fcT,   // [B,G,Q]  Cm rows
                      const int*   __restrict__ nactual,
                      float* __restrict__ out_inds,    // [B,Q]
                      float* __restrict__ out_mask)    // [B,Q]
{
    __shared__ float v_s[Q_ + 1];
    __shared__ float u_s[G_ + 1];
    __shared__ int   p_s[Q_ + 1];
    __shared__ unsigned char used_s[Q_ + 1];
    __shared__ float red_val[1024];
    __shared__ int   red_idx[1024];

    const int b   = blockIdx.x;
    const int tid = threadIdx.x;

    // Output defaults (d_out is poisoned) + dual init.
    for (int j = tid; j < Q_; j += 1024) {
        out_inds[b * Q_ + j] = 0.0f;
        out_mask[b * Q_ + j] = 0.0f;
    }
    for (int j = tid; j <= Q_; j += 1024) { v_s[j] = 0.0f; p_s[j] = 0; }
    if (tid <= G_) u_s[tid] = 0.0f;
    __syncthreads();

    int n = nactual[b];
    if (n < 0)  n = 0;
    if (n > G_) n = G_;

    for (int i = 1; i <= n; ++i) {
        if (tid == 0) p_s[0] = i;
        for (int j = tid; j <= Q_; j += 1024) used_s[j] = 0;
        __syncthreads();

        int j0 = 0;
        for (;;) {
            if (tid == 0) used_s[j0] = 1;
            __syncthreads();

            const int   i0  = p_s[j0];
            const float ui0 = u_s[i0];
            const float* Crow = fcT + ((size_t)b * G_ + (i0 - 1)) * Q_;

            // Parallel argmin of reduced cost over free columns.
            float bv = INFINITY; int bj = 0x7FFFFFFF;
            for (int j = tid + 1; j <= Q_; j += 1024) {
                if (!used_s[j]) {
                    float cur = Crow[j - 1] - ui0 - v_s[j];
                    if (cur < bv) { bv = cur; bj = j; }   // strict: keeps lowest j
                }
            }
            red_val[tid] = bv; red_idx[tid] = bj;
            __syncthreads();
            #pragma unroll
            for (int s = 512; s > 0; s >>= 1) {
                if (tid < s) {
                    float v2 = red_val[tid + s]; int j2 = red_idx[tid + s];
                    if (v2 < red_val[tid] ||
                        (v2 == red_val[tid] && j2 < red_idx[tid])) {
                        red_val[tid] = v2; red_idx[tid] = j2;
                    }
                }
                __syncthreads();
            }
            const float delta = red_val[0];
            const int   jbest = red_idx[0];

            // u[p[used]] += delta ; v[used] -= delta   (p values distinct)
            for (int j = tid; j <= Q_; j += 1024) {
                if (used_s[j]) {
                    v_s[j] -= delta;
                    u_s[p_s[j]] += delta;
                }
            }
            __syncthreads();

            j0 = jbest;
            if (p_s[j0] == 0) break;   // uniform: jbest comes from shared mem
        }
        if (tid == 0) p_s[j0] = i;     // degenerate augmentation (way[] == 0)
        __syncthreads();
    }

    // rows = p[1:]; matched columns j-1 get gt index p[j]-1, mask 1.
    for (int j = tid + 1; j <= Q_; j += 1024) {
        int r = p_s[j];
        if (r > 0) {
            out_inds[b * Q_ + (j - 1)] = (float)(r - 1);
            out_mask[b * Q_ + (j - 1)] = 1.0f;
        }
    }
}

// ---------------------------------------------------------------------------
extern "C" void kernel_launch(void* const* d_in, const int* in_sizes, int n_in,
                              void* d_out, int out_size, void* d_ws, size_t ws_size,
                              hipStream_t stream) {
    (void)in_sizes; (void)n_in; (void)out_size; (void)ws_size;

    const float* sem    = (const float*)d_in[0];   // [B,Q,C]
    const float* center = (const float*)d_in[1];   // [B,Q,G]
    const float* size_  = (const float*)d_in[2];   // [B,Q,G]
    const float* gious  = (const float*)d_in[3];   // [B,Q,G]
    const int*   labels = (const int*)d_in[4];     // [B,G]
    const int*   nact   = (const int*)d_in[5];     // [B]

    float* out      = (float*)d_out;
    float* out_inds = out;                 // [B,Q]
    float* out_mask = out + B_ * Q_;       // [B,Q]
    float* fc_out   = out + 2 * B_ * Q_;   // [B,Q,G]
    float* fcT      = (float*)d_ws;        // [B,G,Q] = 8 MB scratch

    dim3 grid1(Q_ / 16, B_);
    cost_wmma_kernel<<<grid1, 256, 0, stream>>>(sem, center, size_, gious,
                                                labels, fc_out, fcT);
    hungarian_kernel<<<B_, 1024, 0, stream>>>(fcT, nact, out_inds, out_mask);
}